// MonotonicLocationSensitiveAttention_22136261443626
// MI455X (gfx1250) — compile-verified
//
#include <hip/hip_runtime.h>
#include <hip/hip_bf16.h>
#include <cstddef>

// Problem constants (match reference)
#define BB       64
#define TT       2048
#define RNN_DIM  1024
#define EMB_DIM  512
#define ATT_DIM  128
#define N_FILT   32
#define KSIZE    31

typedef float v2f __attribute__((ext_vector_type(2)));
typedef float v8f __attribute__((ext_vector_type(8)));

// ---------------------------------------------------------------------------
// Kernel 1: pq = q @ Wq^T   (M=64, N=128, K=1024) via V_WMMA_F32_16X16X4_F32
// One wave per 16x16 output tile. A-operand lane layout: m = lane&15,
// a.x = A[m][2*hi], a.y = A[m][2*hi+1] (hi = lane>>4); B mirrored with n.
// ---------------------------------------------------------------------------
__global__ __launch_bounds__(32)
void pq_wmma_kernel(const float* __restrict__ q,    // (64,1024)
                    const float* __restrict__ Wq,   // (128,1024) row-major
                    float* __restrict__ pq)         // (64,128)
{
    const int lane = threadIdx.x;
    const int lo = lane & 15, hi = lane >> 4;
    const int mt = blockIdx.x >> 3;   // 0..3  (M tile)
    const int nt = blockIdx.x & 7;    // 0..7  (N tile)

    // pq[m][n] = sum_k q[m][k] * Wq[n][k]  -> B[k][n] = Wq[n][k]
    const float* arow = q  + (size_t)(mt * 16 + lo) * RNN_DIM;
    const float* brow = Wq + (size_t)(nt * 16 + lo) * RNN_DIM;

    v8f acc = {};
    #pragma unroll 4
    for (int kk = 0; kk < RNN_DIM; kk += 4) {
        v2f a = *(const v2f*)(arow + kk + 2 * hi);  // q[m][kk+2hi .. +1]
        v2f b = *(const v2f*)(brow + kk + 2 * hi);  // Wq[n][kk+2hi .. +1]
        acc = __builtin_amdgcn_wmma_f32_16x16x4_f32(
            false, a, false, b, (short)0, acc, false, false);
    }
    // C/D layout: VGPR v -> M = v + 8*hi, N = lo
    #pragma unroll
    for (int v = 0; v < 8; ++v) {
        const int m = mt * 16 + hi * 8 + v;
        const int n = nt * 16 + lo;
        pq[m * ATT_DIM + n] = acc[v];
    }
}

// ---------------------------------------------------------------------------
// Kernel 2: fused conv1d(2->32,k=31) + loc = conv @ Wloc^T (WMMA, K=32) +
//           e = Wv . tanh(pq + loc + pm) + r + noise, mask -> alignment[b][t]
// One wave per (b, 16-wide t tile). conv features are computed directly into
// the WMMA A-operand lane layout; Wloc/Wconv staged in LDS.
// ---------------------------------------------------------------------------
__global__ __launch_bounds__(32)
void energies_kernel(const float* __restrict__ aw_cat,   // (B,2,T)
                     const float* __restrict__ pm,       // (B,T,128)
                     const float* __restrict__ pq,       // (B,128)
                     const float* __restrict__ Wconv,    // (32,2,31)
                     const float* __restrict__ Wloc,     // (128,32)
                     const float* __restrict__ Wv,       // (128)
                     const float* __restrict__ rp,       // (1)
                     const float* __restrict__ noise,    // (B,T)
                     const unsigned char* __restrict__ mask, // (B,T) bool
                     float* __restrict__ align_out)      // (B,T)
{
    __shared__ float aw_s[2][48];                       // input window t0-15..t0+30
    __shared__ float wconv_s[N_FILT * 2 * KSIZE];       // 1984
    __shared__ float wloc_s[ATT_DIM * N_FILT];          // 4096
    __shared__ float pq_s[ATT_DIM];
    __shared__ float e_s[16];

    const int lane = threadIdx.x;
    const int lo = lane & 15, hi = lane >> 4;
    const int b  = blockIdx.x;
    const int t0 = blockIdx.y * 16;

    for (int i = lane; i < N_FILT * 2 * KSIZE; i += 32) wconv_s[i] = Wconv[i];
    for (int i = lane; i < ATT_DIM * N_FILT;  i += 32) wloc_s[i]  = Wloc[i];
    for (int i = lane; i < ATT_DIM;           i += 32) pq_s[i]    = pq[b * ATT_DIM + i];
    for (int i = lane; i < 46; i += 32) {
        const int t = t0 - 15 + i;
        float v0 = 0.f, v1 = 0.f;
        if (t >= 0 && t < TT) {
            v0 = aw_cat[(size_t)b * 2 * TT + t];
            v1 = aw_cat[(size_t)b * 2 * TT + TT + t];
        }
        aw_s[0][i] = v0;
        aw_s[1][i] = v1;
    }
    __syncthreads();

    // conv features for this lane: row m = lo, filters f = 4*j + 2*hi (+1)
    const int m = lo;
    v2f convA[8];
    #pragma unroll
    for (int j = 0; j < 8; ++j) {
        const int f0 = 4 * j + 2 * hi;
        float a0 = 0.f, a1 = 0.f;
        #pragma unroll
        for (int tau = 0; tau < KSIZE; ++tau) {
            const float x0 = aw_s[0][m + tau];
            const float x1 = aw_s[1][m + tau];
            a0 = fmaf(x0, wconv_s[(f0 * 2 + 0) * KSIZE + tau],
                 fmaf(x1, wconv_s[(f0 * 2 + 1) * KSIZE + tau], a0));
            a1 = fmaf(x0, wconv_s[((f0 + 1) * 2 + 0) * KSIZE + tau],
                 fmaf(x1, wconv_s[((f0 + 1) * 2 + 1) * KSIZE + tau], a1));
        }
        convA[j][0] = a0;
        convA[j][1] = a1;
    }

    float esum[8];
    #pragma unroll
    for (int v = 0; v < 8; ++v) esum[v] = 0.f;

    // loc tile = convA(16x32) @ Wloc^T(32x128), 8 N-tiles x 8 K-steps of WMMA
    for (int ntile = 0; ntile < 8; ++ntile) {
        const int na = ntile * 16 + lo;          // this lane's output column a
        v8f acc = {};
        #pragma unroll
        for (int j = 0; j < 8; ++j) {
            // B[k][n] = Wloc[n][k]; lane supplies k = 4j+2hi, 4j+2hi+1
            v2f bv = *(const v2f*)&wloc_s[na * N_FILT + 4 * j + 2 * hi];
            acc = __builtin_amdgcn_wmma_f32_16x16x4_f32(
                false, convA[j], false, bv, (short)0, acc, false, false);
        }
        const float wv  = Wv[na];
        const float pqa = pq_s[na];
        #pragma unroll
        for (int v = 0; v < 8; ++v) {
            const int t = t0 + hi * 8 + v;       // C layout: M = v + 8*hi
            const float val = acc[v] + pqa +
                pm[((size_t)b * TT + t) * ATT_DIM + na];
            esum[v] += tanhf(val) * wv;
        }
    }

    // reduce over the 16 'a' lanes inside each half-wave
    #pragma unroll
    for (int v = 0; v < 8; ++v) {
        float s = esum[v];
        s += __shfl_xor(s, 1, 32);
        s += __shfl_xor(s, 2, 32);
        s += __shfl_xor(s, 4, 32);
        s += __shfl_xor(s, 8, 32);
        esum[v] = s;
    }
    if (lo == 0) {
        #pragma unroll
        for (int v = 0; v < 8; ++v) e_s[hi * 8 + v] = esum[v];
    }
    __syncthreads();

    if (lane < 16) {
        const int t = t0 + lane;
        float al = e_s[lane] + rp[0] + noise[(size_t)b * TT + t];
        if (mask[(size_t)b * TT + t]) al = -1e20f;
        align_out[(size_t)b * TT + t] = al;
    }
}

// ---------------------------------------------------------------------------
// Kernel 3: per-row monotonic recurrence via two block-level scans.
// One block per batch row; 256 threads x 8 contiguous elements each.
// ---------------------------------------------------------------------------
__global__ __launch_bounds__(256)
void scan_alpha_kernel(const float* __restrict__ align,   // (B,T)
                       const float* __restrict__ aw_cat,  // (B,2,T)
                       float* __restrict__ alpha)         // (B,T)
{
    __shared__ float tot[256];
    const int b = blockIdx.x;
    const int tid = threadIdx.x;
    const int base = tid * 8;

    float p[8], lcp[8], s[8];

    // scan 1: cumsum(log(clip(1-p)))
    float run = 0.f;
    #pragma unroll
    for (int i = 0; i < 8; ++i) {
        const float a = align[(size_t)b * TT + base + i];
        const float pv = 1.f / (1.f + expf(-a));
        p[i] = pv;
        float om = fminf(fmaxf(1.f - pv, 1e-10f), 1.f);
        run += logf(om);
        lcp[i] = run;
    }
    tot[tid] = run;
    __syncthreads();
    if (tid == 0) {
        float acc = 0.f;
        for (int k = 0; k < 256; ++k) { const float t = tot[k]; tot[k] = acc; acc += t; }
    }
    __syncthreads();
    const float off1 = tot[tid];

    // scan 2: cumsum(exp(log_prev - lcp))
    float run2 = 0.f;
    #pragma unroll
    for (int i = 0; i < 8; ++i) {
        lcp[i] += off1;
        const float aw0 = aw_cat[(size_t)b * 2 * TT + base + i];
        const float lp  = logf(fmaxf(aw0, 1e-10f));
        run2 += expf(lp - lcp[i]);
        s[i] = run2;
    }
    __syncthreads();
    tot[tid] = run2;
    __syncthreads();
    if (tid == 0) {
        float acc = 0.f;
        for (int k = 0; k < 256; ++k) { const float t = tot[k]; tot[k] = acc; acc += t; }
    }
    __syncthreads();
    const float off2 = tot[tid];

    #pragma unroll
    for (int i = 0; i < 8; ++i) {
        alpha[(size_t)b * TT + base + i] = p[i] * expf(lcp[i]) * (s[i] + off2);
    }
}

// ---------------------------------------------------------------------------
// Kernel 4/5: ctx[b][d] = sum_t alpha[b][t] * memory[b][t][d]
// 256 MB stream split over (b, 16 t-chunks) blocks; f32 atomics to combine.
// ---------------------------------------------------------------------------
__global__ __launch_bounds__(256)
void zero_kernel(float* __restrict__ p, int n)
{
    const int i = blockIdx.x * 256 + threadIdx.x;
    if (i < n) p[i] = 0.f;
}

__global__ __launch_bounds__(256)
void context_kernel(const float* __restrict__ memory,  // (B,T,512)
                    const float* __restrict__ alpha,   // (B,T)
                    float* __restrict__ ctx)           // (B,512)
{
    __shared__ float al_s[128];
    const int b   = blockIdx.x;
    const int t0  = blockIdx.y * 128;
    const int tid = threadIdx.x;

    if (tid < 128) al_s[tid] = alpha[(size_t)b * TT + t0 + tid];
    __syncthreads();

    const float* mbase = memory + ((size_t)b * TT + t0) * EMB_DIM;
    float acc0 = 0.f, acc1 = 0.f;
    #pragma unroll 4
    for (int t = 0; t < 128; ++t) {
        const float av = al_s[t];
        const float* mrow = mbase + (size_t)t * EMB_DIM;
        if ((t & 7) == 0) {
            // hint the streaming pattern -> global_prefetch_b8
            __builtin_prefetch(mrow + 8 * EMB_DIM + tid, 0, 1);
        }
        acc0 = fmaf(av, mrow[tid], acc0);
        acc1 = fmaf(av, mrow[tid + 256], acc1);
    }
    atomicAdd(&ctx[b * EMB_DIM + tid], acc0);
    atomicAdd(&ctx[b * EMB_DIM + tid + 256], acc1);
}

// ---------------------------------------------------------------------------
// Host launcher
// ---------------------------------------------------------------------------
extern "C" void kernel_launch(void* const* d_in, const int* in_sizes, int n_in,
                              void* d_out, int out_size, void* d_ws, size_t ws_size,
                              hipStream_t stream)
{
    (void)in_sizes; (void)n_in; (void)out_size; (void)ws_size;

    const float* q      = (const float*)d_in[0];   // (64,1024)
    const float* memory = (const float*)d_in[1];   // (64,2048,512)
    const float* pm     = (const float*)d_in[2];   // (64,2048,128)
    const float* awcat  = (const float*)d_in[3];   // (64,2,2048)
    const unsigned char* mask = (const unsigned char*)d_in[4]; // (64,2048) bool
    const float* noise  = (const float*)d_in[5];   // (64,2048)
    const float* Wq     = (const float*)d_in[6];   // (128,1024)
    const float* Wv     = (const float*)d_in[7];   // (1,128)
    const float* Wconv  = (const float*)d_in[8];   // (32,2,31)
    const float* Wloc   = (const float*)d_in[9];   // (128,32)
    const float* r      = (const float*)d_in[10];  // (1)

    // d_out = [attention_context (64*512), alpha (64*2048)] flat f32
    float* ctx   = (float*)d_out;
    float* alpha = (float*)d_out + BB * EMB_DIM;

    // workspace: pq (64*128) then alignments (64*2048)  (~544 KB)
    float* pq_ws    = (float*)d_ws;
    float* align_ws = pq_ws + BB * ATT_DIM;

    pq_wmma_kernel<<<32, 32, 0, stream>>>(q, Wq, pq_ws);
    energies_kernel<<<dim3(BB, TT / 16), 32, 0, stream>>>(
        awcat, pm, pq_ws, Wconv, Wloc, Wv, r, noise, mask, align_ws);
    scan_alpha_kernel<<<BB, 256, 0, stream>>>(align_ws, awcat, alpha);
    zero_kernel<<<(BB * EMB_DIM + 255) / 256, 256, 0, stream>>>(ctx, BB * EMB_DIM);
    context_kernel<<<dim3(BB, TT / 128), 256, 0, stream>>>(memory, alpha, ctx);
}